// SimpleReverb_73160472920644
// MI455X (gfx1250) — compile-verified
//
#include <hip/hip_runtime.h>

// SimpleReverb for MI455X (gfx1250, wave32).
//
// Kernel 1: dry mix + early reflections (pure streaming, memory-bound).
// Kernel 2: 4 feedback combs with one-pole lowpass. The damping IIR is a
//   serial scan; we block 16 timesteps and compute the in-block scan as a
//   lower-triangular 16x16 fp32 matmul (chain of 4x V_WMMA_F32_16X16X4_F32),
//   with 16 batches as the WMMA N dimension. One wave per comb, one
//   workgroup per 16-batch group; comb outputs are summed through LDS
//   (no atomics -> deterministic).
//
// This revision removes the EXEC-divergent predicated loads (they forced
// 8 serialized s_wait_loadcnt 0 round-trips per block) in favor of
// clamped-address branchless loads, and software-pipelines both the
// x/history loads and the out read-modify-write one block ahead, plus
// global_prefetch_b8 further ahead. The per-block critical path is now
// one FMA + one lane shuffle (the scan carry) with memory overlapped.
//
// Workspace: filtered-history circular buffers,
//   4 combs * 4 batch-groups * 8192 slots * 16 batches * 4 B = 8 MB.

typedef __attribute__((ext_vector_type(2))) float v2f;
typedef __attribute__((ext_vector_type(8))) float v8f;

#define T_LEN   65536
#define B_BATCH 64
#define BUF_SZ  8192
#define BUF_MASK (BUF_SZ - 1)

// ---------------------------------------------------------------------------
// Kernel 1: out = x*(1-mix) + (mix/9) * sum_k g_k * delay(x, d_k)
// ---------------------------------------------------------------------------
__global__ void early_dry_kernel(const float* __restrict__ x,
                                 const float* __restrict__ mixp,
                                 float* __restrict__ out)
{
    const int i = blockIdx.x * blockDim.x + threadIdx.x;   // grid sized exactly
    const int t = i & (T_LEN - 1);                         // T is a power of 2
    const float mix  = *mixp;
    const float wetc = mix * (1.0f / 9.0f);

    const int   d[5] = {1014, 1631, 2337, 3131, 3924};
    const float g[5] = {0.8f, 0.6f, 0.4f, 0.3f, 0.2f};

    float acc = 0.0f;
#pragma unroll
    for (int k = 0; k < 5; ++k) {
        const int tm = t - d[k];
        const int tc = tm < 0 ? 0 : tm;                    // clamp, mask below
        const float w = (tm < 0) ? 0.0f : g[k];
        acc += w * x[i - t + tc];                          // branchless tap
    }
    out[i] = x[i] * (1.0f - mix) + wetc * acc;
}

// ---------------------------------------------------------------------------
// Kernel 2: feedback combs via blocked scan (WMMA f32 16x16x4 chain).
//   grid = 4 (batch groups of 16), block = 128 (4 waves, one comb each).
// ---------------------------------------------------------------------------
__global__ void __launch_bounds__(128, 1)
reverb_combs_kernel(const float* __restrict__ x,
                    const float* __restrict__ decayp,
                    const float* __restrict__ dampp,
                    const float* __restrict__ mixp,
                    float* __restrict__ out,
                    float* __restrict__ hist)
{
    __shared__ float S[4][256];          // per-comb 16(time) x 16(batch) tiles

    const int tid  = threadIdx.x;
    const int lane = tid & 31;
    const int comb = tid >> 5;           // one wave32 per comb
    const int bg   = blockIdx.x;         // 16-batch group
    const int n    = lane & 15;          // matrix column: batch within group
    const int hi   = lane >> 4;          // lane half selects K / row stripes

    const float damping = *dampp;
    const float omd     = 1.0f - damping;
    const float decay   = *decayp;
    const float wetc    = (*mixp) * (1.0f / 9.0f);

    const int   Dtab[4]  = {4983, 5600, 6570, 7364};
    const float fbtab[4] = {0.70f, 0.65f, 0.60f, 0.55f};
    const int   D  = Dtab[comb];
    const float fb = fbtab[comb] * decay;

    // powers of damping, p[k] = damping^k
    float p[17];
    p[0] = 1.0f;
#pragma unroll
    for (int k = 1; k <= 16; ++k) p[k] = p[k - 1] * damping;

    // A operand: L[j][i] = damping^(j-i) for i<=j else 0, as four 16x4 slabs.
    // A layout (16x4 f32): lane row M = lane&15; .x -> K = 2*hi, .y -> K = 2*hi+1.
    v2f A[4];
#pragma unroll
    for (int kk = 0; kk < 4; ++kk) {
        const int c0 = 4 * kk + 2 * hi;
        A[kk].x = (c0     <= n) ? p[n - c0]     : 0.0f;
        A[kk].y = (c0 + 1 <= n) ? p[n - c0 - 1] : 0.0f;
    }

    const long  xrow = (long)(bg * 16 + n) * T_LEN;
    float* __restrict__ H = hist + (size_t)(comb * 4 + bg) * BUF_SZ * 16;

    float yprev = 0.0f;                  // filtered[t0-1] carry-in (this column)

    // Branchless load of one block's delayed-path inputs (x and history).
    // q -> (kk = q>>1, half = q&1), row r = 4*kk + 2*hi + (q&1).
    auto issue = [&](int t0, float* xv, float* hv) {
#pragma unroll
        for (int q = 0; q < 8; ++q) {
            const int r  = 4 * (q >> 1) + 2 * hi + (q & 1);
            const int tm = t0 + r - D;
            const int tc = tm < 0 ? 0 : tm;              // clamp, masked later
            xv[q] = x[xrow + tc];
            hv[q] = H[(tc & BUF_MASK) * 16 + n];
        }
        // pull data for ~8 blocks ahead from L2 into the WGP cache
        const int tp  = t0 + 128 - D;
        const int tpc = tp < 0 ? 0 : tp;
        __builtin_prefetch(&x[xrow + tpc], 0, 0);
        __builtin_prefetch(&H[(tpc & BUF_MASK) * 16 + n], 0, 0);
    };

    // Process one 16-timestep block whose inputs are already in registers.
    auto process = [&](int t0, const float* xv, const float* hv, float* ov) {
        // U[r][n] = (1-d) * (x[t-D] + fb*filtered[t-D]),  0 for t < D
        float vals[8];
#pragma unroll
        for (int q = 0; q < 8; ++q) {
            const int r  = 4 * (q >> 1) + 2 * hi + (q & 1);
            const int tm = t0 + r - D;
            const float w = (tm < 0) ? 0.0f : omd;
            vals[q] = w * (xv[q] + fb * hv[q]);
        }

        v8f acc = {0.f, 0.f, 0.f, 0.f, 0.f, 0.f, 0.f, 0.f};
#pragma unroll
        for (int kk = 0; kk < 4; ++kk) {
            v2f Bop;
            Bop.x = vals[2 * kk];
            Bop.y = vals[2 * kk + 1];
            // D(16x16) += A_kk(16x4) @ B_kk(4x16)
            acc = __builtin_amdgcn_wmma_f32_16x16x4_f32(
                      false, A[kk], false, Bop, (short)0, acc, false, false);
        }

        // add carry, emit filtered values to history + LDS tile
        float y[8];
#pragma unroll
        for (int vv = 0; vv < 8; ++vv) {
            const int m = vv + 8 * hi;                   // time row in block
            y[vv] = acc[vv] + p[m + 1] * yprev;
            H[((t0 + m) & BUF_MASK) * 16 + n] = y[vv];
            S[comb][m * 16 + n] = y[vv];
        }
        // next carry = filtered at m=15, held by lane 16+n (vv=7, hi=1)
        yprev = __shfl(y[7], 16 + n, 32);

        __syncthreads();
        // sum the 4 combs; ov[] holds out values preloaded one block ago.
        const int t1 = (t0 + 16 < T_LEN) ? (t0 + 16) : t0;   // next out block
#pragma unroll
        for (int u = 0; u < 2; ++u) {
            const int e  = tid + 128 * u;
            const int j  = e >> 4;
            const int nb = e & 15;
            const float rsum = S[0][e] + S[1][e] + S[2][e] + S[3][e];
            const long  row  = (long)(bg * 16 + nb) * T_LEN;
            out[row + t0 + j] = ov[u] + wetc * rsum;
            ov[u] = out[row + t1 + j];                   // preload next block
        }
        __syncthreads();
    };

    // ---- software-pipelined main loop (2 blocks in flight) -----------------
    float xa[8], ha[8], xb[8], hb[8];
    issue(0, xa, ha);

    float ov[2];
#pragma unroll
    for (int u = 0; u < 2; ++u) {
        const int e  = tid + 128 * u;
        const int j  = e >> 4;
        const int nb = e & 15;
        ov[u] = out[(long)(bg * 16 + nb) * T_LEN + j];
    }

    for (int t0 = 0; t0 < T_LEN; t0 += 32) {
        issue(t0 + 16, xb, hb);                          // overlap with block A
        process(t0, xa, ha, ov);
        const int tn = (t0 + 32 < T_LEN) ? (t0 + 32) : (t0 + 16);
        issue(tn, xa, ha);                               // overlap with block B
        process(t0 + 16, xb, hb, ov);
    }
}

// ---------------------------------------------------------------------------
extern "C" void kernel_launch(void* const* d_in, const int* in_sizes, int n_in,
                              void* d_out, int out_size, void* d_ws, size_t ws_size,
                              hipStream_t stream)
{
    const float* x       = (const float*)d_in[0];
    const float* decay   = (const float*)d_in[1];
    const float* damping = (const float*)d_in[2];
    const float* mix     = (const float*)d_in[3];
    float* out  = (float*)d_out;
    float* hist = (float*)d_ws;          // needs 8 MB

    early_dry_kernel<<<(B_BATCH * T_LEN) / 256, 256, 0, stream>>>(x, mix, out);
    reverb_combs_kernel<<<4, 128, 0, stream>>>(x, decay, damping, mix, out, hist);
}